// Bezier_12240656793685
// MI455X (gfx1250) — compile-verified
//
#include <hip/hip_runtime.h>
#include <math.h>

// CDNA5 / gfx1250: wave32, WMMA 16x16 shapes.
typedef __attribute__((ext_vector_type(2))) float v2f;
typedef __attribute__((ext_vector_type(8))) float v8f;

#define RES   512
#define STEPS 128

// Grid: 1024 blocks = 64 (b*s) pairs * 16 chunks. Block: 256 threads = 8 waves.
// Each wave: one 16-row tile strip (fixed pT), 8 column tiles of 16x16.
// raster[bs][p][q] = sum_t exp(-(y[t]-p/512)^2*inv) * exp(-(x[t]-q/512)^2*inv)
//                  = (Ay^T * Ax)[p][q],  K = 128, done as 32x V_WMMA_F32_16X16X4_F32.
//
// Gaussian recipe (2 VALU + 1 TRANS per element):
//   s  = sqrt(log2(e)/(2*sigma^2))        (uniform)
//   d  = fma(x, s, -g*s)                  (v_fma_f32 / dual-issue)
//   e  = exp2(-(d*d))                     (v_mul + v_exp_f32 with free neg modifier)
// == exp(-(x-g)^2 / (2*sigma^2)) exactly as the reference.
__launch_bounds__(256)
__global__ void bezier_raster_wmma(const float* __restrict__ cps,
                                   const float* __restrict__ sigma_p,
                                   float* __restrict__ out) {
    __shared__ float sx[STEPS];
    __shared__ float sy[STEPS];

    const int tid   = threadIdx.x;
    const int bs    = blockIdx.x >> 4;   // 0..63 : b*4 + s
    const int chunk = blockIdx.x & 15;   // 16 chunks of 64 tiles each

    const float sigma   = sigma_p[0];
    const float invL2E  = (1.0f / (2.0f * sigma * sigma)) * 1.44269504088896341f;
    const float sscale  = sqrtf(invL2E);

    // ---- Phase 1: evaluate cubic Bezier at 128 steps into LDS ----
    // feat[k,t] = C(3,k) * t^(3-k) * (1-t)^k  (reference's _bernstein ordering)
    if (tid < STEPS) {
        const float t  = (float)tid * (1.0f / (float)(STEPS - 1));
        const float u  = 1.0f - t;
        const float f0 = t * t * t;
        const float f1 = 3.0f * t * t * u;
        const float f2 = 3.0f * t * u * u;
        const float f3 = u * u * u;
        const float* cp = cps + bs * 8;            // (4 points) x (x,y)
        sx[tid] = cp[0] * f0 + cp[2] * f1 + cp[4] * f2 + cp[6] * f3;
        sy[tid] = cp[1] * f0 + cp[3] * f1 + cp[5] * f2 + cp[7] * f3;
    }
    __syncthreads();

    // ---- Phase 2: WMMA rasterization ----
    const int wave = tid >> 5;
    const int lane = tid & 31;
    const int m    = lane & 15;              // M (for A) / N (for B) index
    const int kOff = (lane >> 4) << 1;       // lanes 0-15 -> K=0,1 ; lanes 16-31 -> K=2,3

    // wave -> (pT, qBlk): 128 waves per bs cover 32 pT x 4 qBlks (8 qT each)
    const int wgl  = chunk * 8 + wave;       // 0..127
    const int pT   = wgl >> 2;               // 0..31
    const int qBlk = wgl & 3;                // 0..3
    const int p0   = pT << 4;

    // -g_p * s, hoisted per wave (g_p = (p0+m)/512)
    const float ngps = -((float)(p0 + m) * (1.0f / (float)RES)) * sscale;

    // Hoist A-matrix (Ay^T) fragments for this 16-row strip: reused by 8 tiles.
    v2f afrag[32];
#pragma unroll
    for (int kb4 = 0; kb4 < 32; ++kb4) {
        const v2f yy  = *(const v2f*)(sy + kb4 * 4 + kOff);   // ds_load_b64
        const float d0 = fmaf(yy.x, sscale, ngps);
        const float d1 = fmaf(yy.y, sscale, ngps);
        v2f a;
        a.x = __builtin_amdgcn_exp2f(-(d0 * d0));
        a.y = __builtin_amdgcn_exp2f(-(d1 * d1));
        afrag[kb4] = a;
    }

    for (int j = 0; j < 8; ++j) {
        const int q0 = (qBlk * 8 + j) << 4;
        // -g_q * s, hoisted per column tile
        const float ngqs = -((float)(q0 + m) * (1.0f / (float)RES)) * sscale;

        v8f acc = {};
#pragma unroll
        for (int kb4 = 0; kb4 < 32; ++kb4) {
            const v2f xx  = *(const v2f*)(sx + kb4 * 4 + kOff); // ds_load_b64
            const float d0 = fmaf(xx.x, sscale, ngqs);
            const float d1 = fmaf(xx.y, sscale, ngqs);
            v2f b;
            b.x = __builtin_amdgcn_exp2f(-(d0 * d0));
            b.y = __builtin_amdgcn_exp2f(-(d1 * d1));
            // D = A(16x4 f32) x B(4x16 f32) + C : exact f32 accumulation
            acc = __builtin_amdgcn_wmma_f32_16x16x4_f32(
                /*neg_a=*/false, afrag[kb4],
                /*neg_b=*/false, b,
                /*c_mod=*/(short)0, acc,
                /*reuse_a=*/false, /*reuse_b=*/false);
        }

        // C/D layout: VGPR r -> row p0 + r + 8*(lane>=16), col q0 + (lane&15)
        const int row0 = p0 + ((lane >> 4) << 3);
        float* o = out + ((size_t)bs * RES + row0) * RES + (q0 + m);
#pragma unroll
        for (int r = 0; r < 8; ++r) {
            o[(size_t)r * RES] = acc[r];
        }
    }
}

extern "C" void kernel_launch(void* const* d_in, const int* in_sizes, int n_in,
                              void* d_out, int out_size, void* d_ws, size_t ws_size,
                              hipStream_t stream) {
    (void)in_sizes; (void)n_in; (void)d_ws; (void)ws_size; (void)out_size;
    const float* cps   = (const float*)d_in[0];   // (16,4,4,2) f32
    const float* sigma = (const float*)d_in[1];   // scalar f32
    float* out = (float*)d_out;                   // (16,4,1,512,512) f32

    dim3 grid(1024);
    dim3 block(256);
    hipLaunchKernelGGL(bezier_raster_wmma, grid, block, 0, stream, cps, sigma, out);
}